// RNN_33964601377372
// MI455X (gfx1250) — compile-verified
//
#include <hip/hip_runtime.h>

// RNN scan on gfx1250 using V_WMMA_F32_16X16X4_F32.
// h_aug = [h(20), x_t, 1, pad] (K=24), W' = augmented (24 x 32) weights:
//   W'[k][j] = W_hh[j][k] (k<20), W_ih[j] (k==20), b_ih[j]+b_hh[j] (k==21), else 0
// Each wave owns TPW independent 16-row batch tiles (independent dependency
// chains: with only B/16 = 256 tiles total, waves sit one-per-SIMD, so
// intra-wave ILP is the only latency hiding available for the serial scan).
// x is software-pipelined 8 steps ahead via float4 double-buffering so HBM
// latency never enters the recurrence chain.

typedef __attribute__((ext_vector_type(2))) float v2f;
typedef __attribute__((ext_vector_type(4))) float v4f;
typedef __attribute__((ext_vector_type(8))) float v8f;

#define HSZ 20
#define S   34      // padded LDS row stride in floats (b64 alignment, bank spread)
#define TPW 4       // 16-row tiles per wave (4 independent chains)

__device__ __forceinline__ float relu(float v) {
    // single v_med3_num_f32 (no canonicalize pair like fmaxf emits)
    return __builtin_amdgcn_fmed3f(v, 0.0f, 3.402823466e38f);
}

__global__ __launch_bounds__(32) void rnn_scan_wmma_f32(
    const float* __restrict__ x,     // [B, T]
    const float* __restrict__ W_ih,  // [20]
    const float* __restrict__ W_hh,  // [20,20] row-major [j][k]
    const float* __restrict__ b_ih,  // [20]
    const float* __restrict__ b_hh,  // [20]
    const float* __restrict__ fc_w,  // [20]
    const float* __restrict__ fc_b,  // [1]
    float* __restrict__ out,         // [B]
    int B, int T)
{
    __shared__ float htile[TPW][16 * S];

    const int lane = threadIdx.x;          // 0..31, one wave per block
    const int m    = lane & 15;
    const int hi   = lane >> 4;
    const int koff = hi * 2;
    const int base = blockIdx.x * (16 * TPW);

    // ---- constant augmented-weight B operand (resident in VGPRs) ----
    auto welem = [&](int k, int j) -> float {
        float v = 0.0f;
        if (j < HSZ) {
            if (k < HSZ)            v = W_hh[j * HSZ + k];
            else if (k == HSZ)      v = W_ih[j];
            else if (k == HSZ + 1)  v = b_ih[j] + b_hh[j];
        }
        return v;
    };
    v2f bw[2][6];
    #pragma unroll
    for (int n = 0; n < 2; ++n) {
        #pragma unroll
        for (int kk = 0; kk < 6; ++kk) {
            const int j  = n * 16 + m;
            const int k0 = kk * 4 + koff;
            bw[n][kk].x = welem(k0, j);
            bw[n][kk].y = welem(k0 + 1, j);
        }
    }

    // ---- zero initial hidden-state tiles ----
    for (int i = lane; i < TPW * 16 * S; i += 32) (&htile[0][0])[i] = 0.0f;
    __syncthreads();

    const float* xrow[TPW];
    #pragma unroll
    for (int tt = 0; tt < TPW; ++tt) {
        int row = base + tt * 16 + m;
        if (row >= B) row = B - 1;                 // clamp (padding rows never stored)
        xrow[tt] = x + (size_t)row * (size_t)T;
    }

    // ---- one recurrence step for all TPW tiles (loads -> WMMAs -> relu/store) ----
    auto step = [&](const float (&xv)[TPW]) {
        v2f a[TPW][6];
        #pragma unroll
        for (int tt = 0; tt < TPW; ++tt) {
            #pragma unroll
            for (int kk = 0; kk < 5; ++kk)
                a[tt][kk] = *(const v2f*)&htile[tt][m * S + kk * 4 + koff]; // ds_load_b64
            a[tt][5].x = hi ? 0.0f : xv[tt];       // K=20 (x) / K=22 (pad)
            a[tt][5].y = hi ? 0.0f : 1.0f;         // K=21 (bias one) / K=23 (pad)
        }
        v8f c[TPW][2];
        #pragma unroll
        for (int tt = 0; tt < TPW; ++tt) {
            c[tt][0] = (v8f){};
            c[tt][1] = (v8f){};
            #pragma unroll
            for (int kk = 0; kk < 6; ++kk) {
                c[tt][0] = __builtin_amdgcn_wmma_f32_16x16x4_f32(
                               false, a[tt][kk], false, bw[0][kk], (short)0,
                               c[tt][0], false, false);
                c[tt][1] = __builtin_amdgcn_wmma_f32_16x16x4_f32(
                               false, a[tt][kk], false, bw[1][kk], (short)0,
                               c[tt][1], false, false);
            }
        }
        #pragma unroll
        for (int tt = 0; tt < TPW; ++tt) {
            #pragma unroll
            for (int r = 0; r < 8; ++r) {
                const int row = hi * 8 + r;        // D layout: row=r+8*hi, col=16n+m
                htile[tt][row * S + m]      = relu(c[tt][0][r]);
                htile[tt][row * S + 16 + m] = relu(c[tt][1][r]);  // cols 20..31 stay 0
            }
        }
    };

    if ((T & 3) == 0) {
        // pipelined path: float4 double buffer, loads issued 8 steps ahead
        const int G = T >> 2;
        v4f xb0[TPW], xb1[TPW];
        #pragma unroll
        for (int tt = 0; tt < TPW; ++tt) {
            xb0[tt] = *(const v4f*)(xrow[tt]);
            xb1[tt] = (G > 1) ? *(const v4f*)(xrow[tt] + 4) : xb0[tt];
        }
        #pragma unroll 1
        for (int g = 0; g < G; ++g) {
            v4f cur[TPW];
            #pragma unroll
            for (int tt = 0; tt < TPW; ++tt) { cur[tt] = xb0[tt]; xb0[tt] = xb1[tt]; }
            const int gg = (g + 2 < G) ? (g + 2) : (G - 1);   // clamped, branch-free
            #pragma unroll
            for (int tt = 0; tt < TPW; ++tt)
                xb1[tt] = *(const v4f*)(xrow[tt] + (size_t)gg * 4);
            #pragma unroll
            for (int u = 0; u < 4; ++u) {
                float xv[TPW];
                #pragma unroll
                for (int tt = 0; tt < TPW; ++tt) xv[tt] = cur[tt][u];
                step(xv);
            }
        }
    } else {
        // generic fallback
        #pragma unroll 1
        for (int t = 0; t < T; ++t) {
            float xv[TPW];
            #pragma unroll
            for (int tt = 0; tt < TPW; ++tt) xv[tt] = xrow[tt][t];
            step(xv);
        }
    }
    __syncthreads();

    // ---- FC head: out[b] = fc_b + sum_j h[b][j] * fc_w[j] ----
    if (lane < 16) {
        #pragma unroll
        for (int tt = 0; tt < TPW; ++tt) {
            const int row = base + tt * 16 + lane;
            if (row < B) {
                float acc = fc_b[0];
                #pragma unroll
                for (int j = 0; j < HSZ; ++j)
                    acc += htile[tt][lane * S + j] * fc_w[j];
                out[row] = acc;
            }
        }
    }
}

extern "C" void kernel_launch(void* const* d_in, const int* in_sizes, int n_in,
                              void* d_out, int out_size, void* d_ws, size_t ws_size,
                              hipStream_t stream) {
    const float* x    = (const float*)d_in[0];
    const float* W_ih = (const float*)d_in[1];
    const float* W_hh = (const float*)d_in[2];
    const float* b_ih = (const float*)d_in[3];
    const float* b_hh = (const float*)d_in[4];
    const float* fc_w = (const float*)d_in[5];
    const float* fc_b = (const float*)d_in[6];
    float* out = (float*)d_out;

    const int B = out_size;                      // output is [B, 1]
    const int T = in_sizes[0] / B;               // x is [B, T, 1]
    const int rowsPerBlock = 16 * TPW;
    const int blocks = (B + rowsPerBlock - 1) / rowsPerBlock;

    rnn_scan_wmma_f32<<<blocks, 32, 0, stream>>>(x, W_ih, W_hh, b_ih, b_hh,
                                                 fc_w, fc_b, out, B, T);
}